// ArapLoss_82652350644346
// MI455X (gfx1250) — compile-verified
//
#include <hip/hip_runtime.h>
#include <math.h>

// ---------------------------------------------------------------------------
// ARAP energy for MI455X (gfx1250, wave32).
// Memory-bound scatter/gather workload; WMMA (v_wmma_f32_16x16x4_f32) is used
// as the wave-wide reduction engine for the scalar loss.
// d_ws layout: S covariances (nV*9 f32) | R rotations (nV*9 f32)  (~36 MB)
// ---------------------------------------------------------------------------

#define LAMBDA_REG_LEN 1e-6f

typedef __attribute__((ext_vector_type(2))) float v2f;
typedef __attribute__((ext_vector_type(8))) float v8f;

struct alignas(16) LL2 { long long x, y; };

// Wave32 sum reduction through the matrix unit:
// A(16x4): lane L<16 supplies A[L][0]=x_L, lane L>=16 supplies A[L-16][2]=x_L.
// B(4x16) = ones  =>  D[m][n] = x_m + x_{m+16} for every n.
// Per-lane sum of the 8 D registers gives half-sums; xor-16 shuffle completes.
__device__ __forceinline__ float wave_reduce_wmma(float x) {
  v2f a; a[0] = x;    a[1] = 0.0f;
  v2f b; b[0] = 1.0f; b[1] = 1.0f;
  v8f c = {};
  v8f d = __builtin_amdgcn_wmma_f32_16x16x4_f32(
      /*neg_a=*/false, a, /*neg_b=*/false, b,
      /*c_mod=*/(short)0, c, /*reuse_a=*/false, /*reuse_b=*/false);
  float s = 0.0f;
#pragma unroll
  for (int k = 0; k < 8; ++k) s += d[k];
  s += __shfl_xor(s, 16, 32);
  return s;  // full wave sum in every lane
}

// ---------------------------------------------------------------------------
// Kernel 0: zero the S accumulator and the output scalar.
// ---------------------------------------------------------------------------
__global__ void arap_init(float* __restrict__ S, float* __restrict__ out, int nS) {
  int i = blockIdx.x * blockDim.x + threadIdx.x;
  if (i < nS) S[i] = 0.0f;
  if (i == 0) out[0] = 0.0f;
}

// ---------------------------------------------------------------------------
// Kernel 1: per-edge outer products scattered to both endpoint covariances.
// 18 global_atomic_add_f32 per edge; all resident in the 192MB L2.
// ---------------------------------------------------------------------------
__global__ void arap_scatter(const float* __restrict__ vt, const float* __restrict__ v0,
                             const long long* __restrict__ nb, float* __restrict__ S,
                             int nE) {
  int e = blockIdx.x * blockDim.x + threadIdx.x;
  if (e >= nE) return;
  // stream the edge list; prefetch well ahead (emits global_prefetch_b8)
  __builtin_prefetch(nb + 2 * e + 4096, 0, 1);

  LL2 p = ((const LL2*)nb)[e];             // one GLOBAL_LOAD_B128
  int i = (int)p.x, j = (int)p.y;

  float dt[3], d0[3];
#pragma unroll
  for (int k = 0; k < 3; ++k) {
    dt[k] = vt[3 * i + k] - vt[3 * j + k];
    d0[k] = v0[3 * i + k] - v0[3 * j + k];
  }
  float* Si = S + 9 * i;
  float* Sj = S + 9 * j;
#pragma unroll
  for (int r = 0; r < 3; ++r)
#pragma unroll
    for (int cI = 0; cI < 3; ++cI) {
      float v = dt[r] * d0[cI];
      atomicAdd(Si + 3 * r + cI, v);
      atomicAdd(Sj + 3 * r + cI, v);
    }
}

// ---------------------------------------------------------------------------
// 3x3 symmetric Jacobi rotation (S <- G^T S G, V <- V G), for the SVD below.
// ---------------------------------------------------------------------------
__device__ __forceinline__ void jacobi_rot(float s[3][3], float v[3][3], int p, int q) {
  float apq = s[p][q];
  if (fabsf(apq) < 1e-18f) return;
  float tau = (s[q][q] - s[p][p]) / (2.0f * apq);
  float t   = copysignf(1.0f, tau) / (fabsf(tau) + sqrtf(1.0f + tau * tau));
  float c   = rsqrtf(1.0f + t * t);
  float sn  = t * c;
#pragma unroll
  for (int k = 0; k < 3; ++k) {  // columns: S <- S G
    float skp = s[k][p], skq = s[k][q];
    s[k][p] = c * skp - sn * skq;
    s[k][q] = sn * skp + c * skq;
  }
#pragma unroll
  for (int k = 0; k < 3; ++k) {  // rows: S <- G^T S
    float spk = s[p][k], sqk = s[q][k];
    s[p][k] = c * spk - sn * sqk;
    s[q][k] = sn * spk + c * sqk;
  }
#pragma unroll
  for (int k = 0; k < 3; ++k) {  // accumulate V <- V G
    float vkp = v[k][p], vkq = v[k][q];
    v[k][p] = c * vkp - sn * vkq;
    v[k][q] = sn * vkp + c * vkq;
  }
}

// ---------------------------------------------------------------------------
// Kernel 2: nearest rotation R of each vertex covariance (polar via Jacobi SVD
// of A^T A, U rebuilt with Gram-Schmidt + cross product so det(R)=+1 exactly,
// matching the reference's sigma-sign fix). Also accumulates the reg term.
// ---------------------------------------------------------------------------
__global__ void arap_rotations(const float* __restrict__ vt, const float* __restrict__ v0,
                               const float* __restrict__ S, float* __restrict__ R,
                               float* __restrict__ out, int nV) {
  __shared__ float blockAcc;
  if (threadIdx.x == 0) blockAcc = 0.0f;
  __syncthreads();

  int vi = blockIdx.x * blockDim.x + threadIdx.x;
  float partial = 0.0f;

  if (vi < nV) {
    float A[3][3];
#pragma unroll
    for (int r = 0; r < 3; ++r)
#pragma unroll
      for (int cI = 0; cI < 3; ++cI) A[r][cI] = S[9 * vi + 3 * r + cI];

    // ATA (symmetric)
    float s[3][3];
#pragma unroll
    for (int p = 0; p < 3; ++p)
#pragma unroll
      for (int q = 0; q < 3; ++q)
        s[p][q] = A[0][p] * A[0][q] + A[1][p] * A[1][q] + A[2][p] * A[2][q];

    float V[3][3] = {{1.f, 0.f, 0.f}, {0.f, 1.f, 0.f}, {0.f, 0.f, 1.f}};
#pragma unroll
    for (int sweep = 0; sweep < 4; ++sweep) {
      jacobi_rot(s, V, 0, 1);
      jacobi_rot(s, V, 0, 2);
      jacobi_rot(s, V, 1, 2);
    }

    // sort eigenpairs descending
    float eig[3] = {s[0][0], s[1][1], s[2][2]};
#pragma unroll
    for (int a = 0; a < 2; ++a)
#pragma unroll
      for (int b2 = a + 1; b2 < 3; ++b2)
        if (eig[a] < eig[b2]) {
          float te = eig[a]; eig[a] = eig[b2]; eig[b2] = te;
#pragma unroll
          for (int k = 0; k < 3; ++k) {
            float tv = V[k][a]; V[k][a] = V[k][b2]; V[k][b2] = tv;
          }
        }

    float vc0[3] = {V[0][0], V[1][0], V[2][0]};
    float vc1[3] = {V[0][1], V[1][1], V[2][1]};

    // u1 = normalize(A v1); u2 = normalize(GS(A v2)); u3 = u1 x u2
    float b1[3], b2v[3], u1[3], u2[3], u3[3], v3[3];
#pragma unroll
    for (int r = 0; r < 3; ++r) {
      b1[r]  = A[r][0] * vc0[0] + A[r][1] * vc0[1] + A[r][2] * vc0[2];
      b2v[r] = A[r][0] * vc1[0] + A[r][1] * vc1[1] + A[r][2] * vc1[2];
    }
    float n1 = b1[0] * b1[0] + b1[1] * b1[1] + b1[2] * b1[2];
    float in1 = rsqrtf(fmaxf(n1, 1e-24f));
#pragma unroll
    for (int r = 0; r < 3; ++r) u1[r] = b1[r] * in1;
    float proj = u1[0] * b2v[0] + u1[1] * b2v[1] + u1[2] * b2v[2];
#pragma unroll
    for (int r = 0; r < 3; ++r) b2v[r] -= proj * u1[r];
    float n2 = b2v[0] * b2v[0] + b2v[1] * b2v[1] + b2v[2] * b2v[2];
    float in2 = rsqrtf(fmaxf(n2, 1e-24f));
#pragma unroll
    for (int r = 0; r < 3; ++r) u2[r] = b2v[r] * in2;
    u3[0] = u1[1] * u2[2] - u1[2] * u2[1];
    u3[1] = u1[2] * u2[0] - u1[0] * u2[2];
    u3[2] = u1[0] * u2[1] - u1[1] * u2[0];
    v3[0] = vc0[1] * vc1[2] - vc0[2] * vc1[1];
    v3[1] = vc0[2] * vc1[0] - vc0[0] * vc1[2];
    v3[2] = vc0[0] * vc1[1] - vc0[1] * vc1[0];

    // R = U V^T  (det(U)=det(V)=+1 by construction -> proper rotation)
#pragma unroll
    for (int r = 0; r < 3; ++r)
#pragma unroll
      for (int cI = 0; cI < 3; ++cI)
        R[9 * vi + 3 * r + cI] = u1[r] * vc0[cI] + u2[r] * vc1[cI] + u3[r] * v3[cI];

    // regularizer: 1e-6 * ||vt - v0||^2
    float dd = 0.0f;
#pragma unroll
    for (int k = 0; k < 3; ++k) {
      float d = vt[3 * vi + k] - v0[3 * vi + k];
      dd += d * d;
    }
    partial = LAMBDA_REG_LEN * dd;
  }

  // EXEC is all-ones here (re-converged) -> WMMA reduction is legal
  float wsum = wave_reduce_wmma(partial);
  if ((threadIdx.x & 31) == 0) atomicAdd(&blockAcc, wsum);  // ds_add_f32
  __syncthreads();
  if (threadIdx.x == 0) atomicAdd(out, blockAcc);           // 1 global atomic/block
}

// ---------------------------------------------------------------------------
// Kernel 3: per-edge ARAP residual, reduced via WMMA + LDS + 1 atomic/block.
// ---------------------------------------------------------------------------
__global__ void arap_residual(const float* __restrict__ vt, const float* __restrict__ v0,
                              const long long* __restrict__ nb, const float* __restrict__ R,
                              float* __restrict__ out, int nE) {
  __shared__ float blockAcc;
  if (threadIdx.x == 0) blockAcc = 0.0f;
  __syncthreads();

  int e = blockIdx.x * blockDim.x + threadIdx.x;
  float partial = 0.0f;

  if (e < nE) {
    __builtin_prefetch(nb + 2 * e + 4096, 0, 1);
    LL2 p = ((const LL2*)nb)[e];
    int i = (int)p.x, j = (int)p.y;

    float dt[3], d0[3];
#pragma unroll
    for (int k = 0; k < 3; ++k) {
      dt[k] = vt[3 * i + k] - vt[3 * j + k];
      d0[k] = v0[3 * i + k] - v0[3 * j + k];
    }
    const float* Ri = R + 9 * i;
    const float* Rj = R + 9 * j;
#pragma unroll
    for (int r = 0; r < 3; ++r) {
      float rot = 0.0f;
#pragma unroll
      for (int cI = 0; cI < 3; ++cI)
        rot += 0.5f * (Ri[3 * r + cI] + Rj[3 * r + cI]) * d0[cI];
      float acc = dt[r] - rot;
      partial += acc * acc;
    }
  }

  float wsum = wave_reduce_wmma(partial);
  if ((threadIdx.x & 31) == 0) atomicAdd(&blockAcc, wsum);
  __syncthreads();
  if (threadIdx.x == 0) atomicAdd(out, blockAcc);
}

// ---------------------------------------------------------------------------
extern "C" void kernel_launch(void* const* d_in, const int* in_sizes, int n_in,
                              void* d_out, int out_size, void* d_ws, size_t ws_size,
                              hipStream_t stream) {
  const float*     vt = (const float*)d_in[0];
  const float*     v0 = (const float*)d_in[1];
  const long long* nb = (const long long*)d_in[2];
  float* out = (float*)d_out;

  const int nV = in_sizes[0] / 3;
  const int nE = in_sizes[2] / 2;
  const int nS = nV * 9;

  float* S = (float*)d_ws;            // nS floats
  float* R = S + (size_t)nS;          // nS floats

  const int B = 256;
  arap_init     <<<(nS + B - 1) / B, B, 0, stream>>>(S, out, nS);
  arap_scatter  <<<(nE + B - 1) / B, B, 0, stream>>>(vt, v0, nb, S, nE);
  arap_rotations<<<(nV + B - 1) / B, B, 0, stream>>>(vt, v0, S, R, out, nV);
  arap_residual <<<(nE + B - 1) / B, B, 0, stream>>>(vt, v0, nb, R, out, nE);
}